// MHDNet_59330678227575
// MI455X (gfx1250) — compile-verified
//
#include <hip/hip_runtime.h>
#include <math.h>

#define B_   8
#define L_   12
#define N_   1024
#define H_   32
#define T_   10
#define ND_  10
#define CDIM 32
#define JDIM 8192          // B_ * CDIM * H_
#define FHDIM 276
#define F1OUT 512

typedef float v2f __attribute__((ext_vector_type(2)));
typedef float v8f __attribute__((ext_vector_type(8)));
typedef int   v4i_async __attribute__((vector_size(16)));
typedef __attribute__((address_space(1))) v4i_async* gas_v4i_p;
typedef __attribute__((address_space(3))) v4i_async* lds_v4i_p;

#if defined(__has_builtin)
#if __has_builtin(__builtin_amdgcn_global_load_async_to_lds_b128)
#define ASYNC_LDS 1
#endif
#if __has_builtin(__builtin_amdgcn_s_wait_asynccnt)
#define HAVE_WAIT_ASYNC 1
#endif
#endif
#ifndef ASYNC_LDS
#define ASYNC_LDS 0
#endif
#ifndef HAVE_WAIT_ASYNC
#define HAVE_WAIT_ASYNC 0
#endif

__device__ __forceinline__ void wait_async0() {
#if HAVE_WAIT_ASYNC
  __builtin_amdgcn_s_wait_asynccnt(0);
#else
  asm volatile("s_wait_asynccnt 0x0" ::: "memory");
#endif
}

__device__ __forceinline__ v8f wmma_f32x4(v2f a, v2f b, v8f c) {
  // D = A(16x4 f32) * B(4x16 f32) + C(16x16 f32)
  return __builtin_amdgcn_wmma_f32_16x16x4_f32(false, a, false, b, (short)0, c, false, false);
}

// ---------------- Stage 1: embeddings + hist ----------------
__global__ void k_embed(const float* __restrict__ hd, const float* __restrict__ W_emb,
                        const float* __restrict__ b_emb, const float* __restrict__ T_emb,
                        const float* __restrict__ D_emb,
                        float* __restrict__ hist, float* __restrict__ tidb, float* __restrict__ diwb) {
  int row = blockIdx.x * blockDim.x + threadIdx.x;
  if (row >= B_ * L_ * N_) return;
  const float* h5 = hd + (size_t)row * 5;
  float f0 = h5[0], f1 = h5[1], f2 = h5[2];
  int ti = (int)(h5[3] * 288.0f); ti = ti < 0 ? 0 : (ti > 287 ? 287 : ti);
  int di = (int)(h5[4]);          di = di < 0 ? 0 : (di > 6 ? 6 : di);
#pragma unroll
  for (int h = 0; h < H_; ++h)
    hist[(size_t)row * H_ + h] = f0 * W_emb[h] + f1 * W_emb[H_ + h] + f2 * W_emb[2 * H_ + h] + b_emb[h];
#pragma unroll
  for (int t = 0; t < T_; ++t) {
    tidb[(size_t)row * T_ + t] = T_emb[ti * T_ + t];
    diwb[(size_t)row * T_ + t] = D_emb[di * T_ + t];
  }
}

// ---------------- Stage 2: sigmoid-gated decomposition xs / h0 ----------------
__global__ void k_std(const float* __restrict__ hist, const float* __restrict__ tidb,
                      const float* __restrict__ diwb, const float* __restrict__ node_u,
                      const float* __restrict__ W_std, const float* __restrict__ b_std,
                      float* __restrict__ h0) {
  int gid = blockIdx.x * blockDim.x + threadIdx.x;
  if (gid >= B_ * L_ * N_ * H_) return;
  int row = gid >> 5;
  int h = gid & 31;
  int n = row & (N_ - 1);
  const float* hrow = hist + (size_t)row * H_;
  const float* trow = tidb + (size_t)row * T_;
  const float* drow = diwb + (size_t)row * T_;
  const float* urow = node_u + n * ND_;
  const float* W0 = W_std + h;              // (62,32) column h, stride 32
  const float* W1 = W_std + 62 * 32 + h;
  float s0 = b_std[h], s1 = b_std[32 + h];
#pragma unroll
  for (int j = 0; j < 32; ++j) { float v = hrow[j]; s0 += v * W0[j * 32]; s1 += v * W1[j * 32]; }
#pragma unroll
  for (int j = 0; j < 10; ++j) { float v = urow[j]; s0 += v * W0[(32 + j) * 32]; s1 += v * W1[(32 + j) * 32]; }
#pragma unroll
  for (int j = 0; j < 10; ++j) { float v = trow[j]; s0 += v * W0[(42 + j) * 32]; s1 += v * W1[(42 + j) * 32]; }
#pragma unroll
  for (int j = 0; j < 10; ++j) { float v = drow[j]; s0 += v * W0[(52 + j) * 32]; s1 += v * W1[(52 + j) * 32]; }
  float hv = hrow[h];
  float x0v = hv / (1.0f + expf(-s0));
  float x1v = hv / (1.0f + expf(-s1));
  float x2v = hv - x0v - x1v;
  float* orow = h0 + (size_t)row * 96;
  orow[h] = x0v; orow[32 + h] = x1v; orow[64 + h] = x2v;
}

// ---------------- Stage 3: ratios for clustering ----------------
__global__ void k_ratio(const float* __restrict__ hist, const float* __restrict__ h0,
                        const float* __restrict__ W_mlp, const float* __restrict__ b_mlp,
                        float* __restrict__ ratio) {
  int gid = blockIdx.x * blockDim.x + threadIdx.x;
  if (gid >= B_ * N_) return;
  int b = gid >> 10, n = gid & 1023;
  float hm = 0.f, x0m = 0.f, x1m = 0.f, x2m = 0.f;
  for (int l = 0; l < L_; ++l) {
    size_t row = (size_t)((b * L_ + l) * N_) + n;
    const float* hr = hist + row * H_;
    const float* xr = h0 + row * 96;
#pragma unroll
    for (int h = 0; h < 32; ++h) {
      hm  += hr[h]      * W_mlp[h];
      x0m += xr[h]      * W_mlp[32 + h];
      x1m += xr[32 + h] * W_mlp[64 + h];
      x2m += xr[64 + h] * W_mlp[96 + h];
    }
  }
  float inv12 = 1.0f / 12.0f;
  float hmean = hm * inv12 + b_mlp[0];
  float d = 1.0f / (hmean + 1e-6f);
  ratio[(size_t)gid * 3 + 0] = (x0m * inv12 + b_mlp[1]) * d;
  ratio[(size_t)gid * 3 + 1] = (x1m * inv12 + b_mlp[2]) * d;
  ratio[(size_t)gid * 3 + 2] = (x2m * inv12 + b_mlp[3]) * d;
}

__global__ void k_centers(const float* __restrict__ ratio, float* __restrict__ centers) {
  __shared__ float sred[256];
  int b = blockIdx.x / 3, p = blockIdx.x % 3;
  float m = -3.4e38f;
  for (int n = threadIdx.x; n < N_; n += 256)
    m = fmaxf(m, ratio[(size_t)(b * N_ + n) * 3 + p]);
  sred[threadIdx.x] = m;
  __syncthreads();
  for (int s = 128; s > 0; s >>= 1) {
    if (threadIdx.x < s) sred[threadIdx.x] = fmaxf(sred[threadIdx.x], sred[threadIdx.x + s]);
    __syncthreads();
  }
  if (threadIdx.x == 0) centers[b * 3 + p] = sred[0];
}

__global__ void k_label(const float* __restrict__ ratio, const float* __restrict__ centers,
                        int* __restrict__ label) {
  int n = blockIdx.x * blockDim.x + threadIdx.x;
  if (n >= N_) return;
  int counts[3] = {0, 0, 0};
  for (int b = 0; b < B_; ++b) {
    float best = 3.4e38f; int bp = 0;
#pragma unroll
    for (int p = 0; p < 3; ++p) {
      float d = fabsf(ratio[(size_t)(b * N_ + n) * 3 + p] - centers[b * 3 + p]);
      if (d < best) { best = d; bp = p; }
    }
    counts[bp]++;
  }
  int lbl = 0, mx = counts[0];
  if (counts[1] > mx) { mx = counts[1]; lbl = 1; }
  if (counts[2] > mx) { lbl = 2; }
  label[n] = lbl;
}

// ---------------- Stage 4: merged normalized adjacency ----------------
__global__ void k_m1m2(const float* __restrict__ node_u, const float* __restrict__ node_d,
                       float* __restrict__ m1, float* __restrict__ m2) {
  int i = blockIdx.x * blockDim.x + threadIdx.x;
  if (i >= N_ * ND_) return;
  m1[i] = tanhf(3.0f * node_u[i]);
  m2[i] = tanhf(3.0f * node_d[i]);
}

__global__ void k_adj(const float* __restrict__ m1, const float* __restrict__ m2,
                      const int* __restrict__ label, float* __restrict__ A) {
  int idx = blockIdx.x * blockDim.x + threadIdx.x;
  if (idx >= N_ * N_) return;
  int nr = idx >> 10, mc = idx & 1023;
  float a = 0.0f;
  if (label[nr] == label[mc]) {
    float d = 0.f;
#pragma unroll
    for (int t = 0; t < ND_; ++t)
      d += m1[nr * ND_ + t] * m2[mc * ND_ + t] - m2[nr * ND_ + t] * m1[mc * ND_ + t];
    a = fmaxf(tanhf(3.0f * d), 0.0f);
  }
  A[(size_t)idx] = a;
}

__global__ void k_rownorm(float* __restrict__ A) {
  __shared__ float sred[256];
  int row = blockIdx.x;
  float s = 0.f;
  for (int i = threadIdx.x; i < N_; i += 256) s += A[(size_t)row * N_ + i];
  sred[threadIdx.x] = s;
  __syncthreads();
  for (int st = 128; st > 0; st >>= 1) {
    if (threadIdx.x < st) sred[threadIdx.x] += sred[threadIdx.x + st];
    __syncthreads();
  }
  float inv = 1.0f / (sred[0] + 1e-6f);
  for (int i = threadIdx.x; i < N_; i += 256) A[(size_t)row * N_ + i] *= inv;
}

// ---------------- Stage 5: x0 in (n, j=(b,c,h)) layout ----------------
__global__ void k_x0(const float* __restrict__ hist, const float* __restrict__ W_start,
                     const float* __restrict__ b_start, float* __restrict__ x0t) {
  size_t gid = (size_t)blockIdx.x * blockDim.x + threadIdx.x;
  if (gid >= (size_t)N_ * JDIM) return;
  int n = (int)(gid >> 13);
  int j = (int)(gid & (JDIM - 1));
  int b = j >> 10;
  int c = (j >> 5) & 31;
  int h = j & 31;
  float s = b_start[c];
#pragma unroll
  for (int l = 0; l < L_; ++l)
    s += W_start[c * L_ + l] * hist[((size_t)((b * L_ + l) * N_) + n) * H_ + h];
  x0t[gid] = s;
}

// ---------------- Stage 6: graph propagation GEMM (WMMA f32 16x16x4) ----------------
// Hout = 0.05*X0 + 0.95 * (A @ Hin), A:(1024x1024), H:(1024x8192)
// grid (16, 64), block 256 (8 waves); wave computes a 16(M) x 64(N) tile.
// The 16-row A panel (64KB) is staged into LDS once per block via async-to-LDS copies
// (shared by all 8 waves), WMMA A-fragments then come from LDS (ds_load_b64).
__global__ void k_prop(const float* __restrict__ A, const float* __restrict__ Hin,
                       const float* __restrict__ X0, float* __restrict__ Hout) {
  extern __shared__ float As[];   // 16 x 1024 floats = 64KB
  int tid = threadIdx.x;
  int mbase = blockIdx.y * 16;
#if ASYNC_LDS
  {
    // 4096 x b128 transfers, 16 per thread; ASYNCcnt-tracked
#pragma unroll 4
    for (int s = 0; s < 16; ++s) {
      int o = tid + 256 * s;        // b128 chunk index
      int r = o >> 8;               // A panel row 0..15
      int cq = o & 255;             // 16B chunk within row
      const float* g = A + (size_t)(mbase + r) * N_ + cq * 4;
      float* l = As + r * N_ + cq * 4;
      __builtin_amdgcn_global_load_async_to_lds_b128(
          (gas_v4i_p)g, (lds_v4i_p)l, 0, 0);
    }
    wait_async0();
  }
#else
  for (int idx = tid; idx < 16 * N_; idx += 256)
    As[idx] = A[(size_t)(mbase + (idx >> 10)) * N_ + (idx & 1023)];
#endif
  __syncthreads();

  int wave = tid >> 5;
  int lane = tid & 31;
  int half = lane >> 4, l16 = lane & 15, koff = half * 2;
  int jbase = blockIdx.x * 512 + wave * 64;
  const float* Arow = As + l16 * N_;      // LDS
  v8f c0 = {}, c1 = {}, c2 = {}, c3 = {};
  for (int k = 0; k < N_; k += 4) {
    v2f a;
    a.x = Arow[k + koff];
    a.y = Arow[k + koff + 1];
    const float* B0 = Hin + (size_t)(k + koff) * JDIM + jbase + l16;
    const float* B1 = B0 + JDIM;
    v2f bb;
    bb.x = B0[0];  bb.y = B1[0];
    c0 = wmma_f32x4(a, bb, c0);
    bb.x = B0[16]; bb.y = B1[16];
    c1 = wmma_f32x4(a, bb, c1);
    bb.x = B0[32]; bb.y = B1[32];
    c2 = wmma_f32x4(a, bb, c2);
    bb.x = B0[48]; bb.y = B1[48];
    c3 = wmma_f32x4(a, bb, c3);
  }
  int rbase = mbase + half * 8;
#pragma unroll
  for (int r = 0; r < 8; ++r) {
    size_t i0 = (size_t)(rbase + r) * JDIM + jbase + l16;
    Hout[i0]      = 0.05f * X0[i0]      + 0.95f * c0[r];
    Hout[i0 + 16] = 0.05f * X0[i0 + 16] + 0.95f * c1[r];
    Hout[i0 + 32] = 0.05f * X0[i0 + 32] + 0.95f * c2[r];
    Hout[i0 + 48] = 0.05f * X0[i0 + 48] + 0.95f * c3[r];
  }
}

// ---------------- Stage 7: fused per-(b,n) tail: W_g, W_rnn, W_e1, W_e2, M, W_f1 (WMMA), W_f2, W_e3 ----------------
__global__ void k_chain(const float* __restrict__ H2, const float* __restrict__ Wg,
                        const float* __restrict__ W_rnn, const float* __restrict__ b_rnn,
                        const float* __restrict__ W_e1, const float* __restrict__ b_e1,
                        const float* __restrict__ W_e2, const float* __restrict__ b_e2,
                        const float* __restrict__ Mm,
                        const float* __restrict__ hist, const float* __restrict__ h0,
                        const float* __restrict__ tidb, const float* __restrict__ diwb,
                        const float* __restrict__ W_f1, const float* __restrict__ b_f1,
                        const float* __restrict__ W_f2, const float* __restrict__ b_f2,
                        const float* __restrict__ W_e3, const float* __restrict__ b_e3,
                        float* __restrict__ out) {
  extern __shared__ char smem[];
  float* Xs  = (float*)smem;                   // 32x128  [0, 16K)
  float* Ys  = (float*)(smem + 16 * 1024);     // 128x128 [16K, 80K)
  float* Gs  = (float*)(smem + 80 * 1024);     // 32x32   [80K, 84K)
  float* FHs = (float*)(smem + 80 * 1024);     // 16x276  [80K, ~97.3K) (reuses Gs)
  float* FC1 = (float*)smem;                   // 16x512  [0, 32K)     (reuses Xs + low Ys)
  float* fcs = (float*)(smem + 98 * 1024);     // 12x3
  int b = blockIdx.x >> 10;
  int n = blockIdx.x & 1023;
  int tid = threadIdx.x;

  // L2-resident weight panels reused by all 8192 blocks
  __builtin_prefetch(W_f1 + (tid << 2), 0, 2);
  __builtin_prefetch(W_f2 + tid, 0, 2);

  // out[b,:,n,:] = H2 row n  @ (Wg0+Wg1)   -> Gs (32x32)
  for (int idx = tid; idx < 32 * 32; idx += 128) {
    int c = idx >> 5, k = idx & 31;
    const float* hrow = H2 + (size_t)n * JDIM + (b * 32 + c) * 32;
    float s = 0.f;
#pragma unroll
    for (int h = 0; h < 32; ++h)
      s += hrow[h] * (Wg[h * 32 + k] + Wg[1024 + h * 32 + k]);
    Gs[idx] = s;
  }
  __syncthreads();

  // X = tanh(Gs @ W_rnn + b_rnn)   (32x128)
  for (int idx = tid; idx < 32 * 128; idx += 128) {
    int c = idx >> 7, hh = idx & 127;
    float s = b_rnn[hh];
#pragma unroll
    for (int h = 0; h < 32; ++h) s += Gs[c * 32 + h] * W_rnn[h * 128 + hh];
    Xs[idx] = tanhf(s);
  }
  __syncthreads();

  // Y = relu(W_e1 @ X + b_e1)   (128x128)
  for (int idx = tid; idx < 128 * 128; idx += 128) {
    int o = idx >> 7, hh = idx & 127;
    float s = b_e1[o];
#pragma unroll
    for (int c = 0; c < 32; ++c) s += W_e1[o * 32 + c] * Xs[c * 128 + hh];
    Ys[idx] = fmaxf(s, 0.0f);
  }
  __syncthreads();

  // FH rows 0..11: [ (W_e2@Y+b_e2)*M | hist | h0 | tid | diw ], relu'ed; rows 12..15 zero pad
  for (int idx = tid; idx < 16 * FHDIM; idx += 128) {
    int row = idx / FHDIM, col = idx % FHDIM;
    float v = 0.0f;
    if (row < 12) {
      size_t r = (size_t)((b * L_ + row) * N_) + n;
      if (col < 128) {
        float s = b_e2[row];
        for (int o = 0; o < 128; ++o) s += W_e2[row * 128 + o] * Ys[o * 128 + col];
        v = s * Mm[(size_t)n * 128 + col];
      } else if (col < 160) v = hist[r * H_ + (col - 128)];
      else if (col < 256)   v = h0[r * 96 + (col - 160)];
      else if (col < 266)   v = tidb[r * T_ + (col - 256)];
      else                  v = diwb[r * T_ + (col - 266)];
      v = fmaxf(v, 0.0f);
    }
    FHs[idx] = v;
  }
  __syncthreads();

  // FC1 = relu(FH(16x276) @ W_f1(276x512) + b_f1) via WMMA f32 16x16x4
  int wave = tid >> 5, lane = tid & 31;
  int half = lane >> 4, l16 = lane & 15, koff = half * 2;
  for (int t = 0; t < 8; ++t) {
    int colb = (wave * 8 + t) * 16;
    v8f c = {};
    for (int k = 0; k < FHDIM; k += 4) {
      v2f a, bb;
      a.x = FHs[l16 * FHDIM + k + koff];
      a.y = FHs[l16 * FHDIM + k + koff + 1];
      bb.x = W_f1[(size_t)(k + koff) * F1OUT + colb + l16];
      bb.y = W_f1[(size_t)(k + koff + 1) * F1OUT + colb + l16];
      c = wmma_f32x4(a, bb, c);
    }
#pragma unroll
    for (int r = 0; r < 8; ++r) {
      int row = r + half * 8;
      int col = colb + l16;
      FC1[row * F1OUT + col] = fmaxf(c[r] + b_f1[col], 0.0f);
    }
  }
  __syncthreads();

  // fc = FC1 @ W_f2 + b_f2  (12x3)
  if (tid < 36) {
    int l = tid / 3, k = tid % 3;
    float s = b_f2[k];
    for (int j = 0; j < F1OUT; ++j) s += FC1[l * F1OUT + j] * W_f2[j * 3 + k];
    fcs[l * 3 + k] = s;
  }
  __syncthreads();

  // fco = W_e3 @ fc + b_e3, write (b, n, o*3+k)
  if (tid < 12) {
    int o = tid / 3, k = tid % 3;
    float s = b_e3[o];
#pragma unroll
    for (int l = 0; l < 12; ++l) s += W_e3[o * 12 + l] * fcs[l * 3 + k];
    out[(size_t)(b * N_ + n) * 12 + o * 3 + k] = s;
  }
}

// ---------------- launcher ----------------
extern "C" void kernel_launch(void* const* d_in, const int* in_sizes, int n_in,
                              void* d_out, int out_size, void* d_ws, size_t ws_size,
                              hipStream_t stream) {
  (void)in_sizes; (void)n_in; (void)out_size; (void)ws_size;
  const float* hd      = (const float*)d_in[0];
  const float* W_emb   = (const float*)d_in[1];
  const float* b_emb   = (const float*)d_in[2];
  const float* T_emb   = (const float*)d_in[3];
  const float* D_emb   = (const float*)d_in[4];
  const float* node_u  = (const float*)d_in[5];
  const float* node_d  = (const float*)d_in[6];
  const float* W_std   = (const float*)d_in[7];
  const float* b_std   = (const float*)d_in[8];
  const float* W_mlp   = (const float*)d_in[9];
  const float* b_mlp   = (const float*)d_in[10];
  const float* W_start = (const float*)d_in[11];
  const float* b_start = (const float*)d_in[12];
  const float* W_g     = (const float*)d_in[13];
  const float* W_rnn   = (const float*)d_in[14];
  const float* b_rnn   = (const float*)d_in[15];
  const float* W_e1    = (const float*)d_in[16];
  const float* b_e1    = (const float*)d_in[17];
  const float* W_e2    = (const float*)d_in[18];
  const float* b_e2    = (const float*)d_in[19];
  const float* Mm      = (const float*)d_in[20];
  const float* W_f1    = (const float*)d_in[21];
  const float* b_f1    = (const float*)d_in[22];
  const float* W_f2    = (const float*)d_in[23];
  const float* b_f2    = (const float*)d_in[24];
  const float* W_e3    = (const float*)d_in[25];
  const float* b_e3    = (const float*)d_in[26];
  float* out = (float*)d_out;
  float* ws = (float*)d_ws;

  // workspace layout (floats)
  const size_t OFF_HIST  = 0;                       // 3,145,728
  const size_t OFF_TID   = OFF_HIST  + 3145728;     //   983,040
  const size_t OFF_DIW   = OFF_TID   + 983040;      //   983,040
  const size_t OFF_H0    = OFF_DIW   + 983040;      // 9,437,184
  const size_t OFF_RATIO = OFF_H0    + 9437184;     //    24,576
  const size_t OFF_CENT  = OFF_RATIO + 24576;       //     1,024
  const size_t OFF_LBL   = OFF_CENT  + 1024;        //     1,024 (int)
  const size_t OFF_M1    = OFF_LBL   + 1024;        //    10,240
  const size_t OFF_M2    = OFF_M1    + 10240;       //    10,240
  const size_t OFF_A     = OFF_M2    + 10240;       // 1,048,576
  const size_t OFF_X0T   = OFF_A     + 1048576;     // 8,388,608
  const size_t OFF_HB1   = OFF_X0T   + 8388608;     // 8,388,608
  const size_t OFF_HB2   = OFF_HB1   + 8388608;     // 8,388,608

  float* hist  = ws + OFF_HIST;
  float* tidb  = ws + OFF_TID;
  float* diwb  = ws + OFF_DIW;
  float* h0    = ws + OFF_H0;
  float* ratio = ws + OFF_RATIO;
  float* cent  = ws + OFF_CENT;
  int*   label = (int*)(ws + OFF_LBL);
  float* m1    = ws + OFF_M1;
  float* m2    = ws + OFF_M2;
  float* Amat  = ws + OFF_A;
  float* x0t   = ws + OFF_X0T;
  float* hb1   = ws + OFF_HB1;
  float* hb2   = ws + OFF_HB2;

  k_embed<<<(B_ * L_ * N_ + 255) / 256, 256, 0, stream>>>(hd, W_emb, b_emb, T_emb, D_emb, hist, tidb, diwb);
  k_std<<<(B_ * L_ * N_ * H_ + 255) / 256, 256, 0, stream>>>(hist, tidb, diwb, node_u, W_std, b_std, h0);
  k_ratio<<<(B_ * N_ + 255) / 256, 256, 0, stream>>>(hist, h0, W_mlp, b_mlp, ratio);
  k_centers<<<B_ * 3, 256, 0, stream>>>(ratio, cent);
  k_label<<<(N_ + 255) / 256, 256, 0, stream>>>(ratio, cent, label);
  k_m1m2<<<(N_ * ND_ + 255) / 256, 256, 0, stream>>>(node_u, node_d, m1, m2);
  k_adj<<<(N_ * N_ + 255) / 256, 256, 0, stream>>>(m1, m2, label, Amat);
  k_rownorm<<<N_, 256, 0, stream>>>(Amat);
  k_x0<<<(unsigned)(((size_t)N_ * JDIM + 255) / 256), 256, 0, stream>>>(hist, W_start, b_start, x0t);

  dim3 pgrid(16, 64);
  size_t asmem = 16 * N_ * sizeof(float);                    // 64KB A panel in LDS
  k_prop<<<pgrid, 256, asmem, stream>>>(Amat, x0t, x0t, hb1); // h1 = 0.05 x0 + 0.95 A@x0
  k_prop<<<pgrid, 256, asmem, stream>>>(Amat, hb1, x0t, hb2); // h2 = 0.05 x0 + 0.95 A@h1

  size_t smem = 98 * 1024 + 256;  // Xs/Ys/Gs/FHs/FC1 regions + fcs
  k_chain<<<B_ * N_, 128, smem, stream>>>(hb2, W_g, W_rnn, b_rnn, W_e1, b_e1, W_e2, b_e2, Mm,
                                          hist, h0, tidb, diwb, W_f1, b_f1, W_f2, b_f2, W_e3, b_e3, out);
}